// RNN_54056458388026
// MI455X (gfx1250) — compile-verified
//
#include <hip/hip_runtime.h>

// ESN forward for MI455X (gfx1250, wave32).
//  - in_proj GEMM  : V_WMMA_F32_16X16X4_F32 tiles (16x16, K=28 = 7 wmma steps)
//  - sequential scan: persistent kernel, W_res (64MB) L2-resident, one wave per
//    reservoir row (coalesced float4), async-to-LDS h staging, global-atomic
//    grid barrier per timestep
//  - readout GEMM  : V_WMMA_F32_16X16X4_F32, N padded 10->16

#define SIZE_IN   28
#define SIZE_RES  4096
#define SIZE_OUT  10
#define BATCH     256
#define TIME      28
#define SEQ       (BATCH * TIME)            // 7168

#define NBLK            128                 // persistent blocks (co-resident)
#define TPB             256                 // 8 waves per block (wave32)
#define ROWS_PER_BLK    (SIZE_RES / NBLK)   // 32
#define WAVES_PER_BLK   (TPB / 32)          // 8
#define ROWS_PER_WAVE   (ROWS_PER_BLK / WAVES_PER_BLK) // 4

typedef float v2f __attribute__((ext_vector_type(2)));
typedef float v8f __attribute__((ext_vector_type(8)));
typedef int   v4i __attribute__((vector_size(16)));   // matches async-LDS builtin

#define WMMA_F32_16X16X4(a, b, c) \
    __builtin_amdgcn_wmma_f32_16x16x4_f32(false, (a), false, (b), (short)0, (c), false, false)

#if __has_builtin(__builtin_amdgcn_global_load_async_to_lds_b128) && \
    __has_builtin(__builtin_amdgcn_s_wait_asynccnt)
#define HAVE_ASYNC_LDS 1
#else
#define HAVE_ASYNC_LDS 0
#endif

// ---------------------------------------------------------------------------
// init: reset barrier counter, load h0 into buffer 0 (graph-replay safe)
// ---------------------------------------------------------------------------
__global__ void esn_init(const float* __restrict__ h0,
                         float* __restrict__ hbuf,
                         unsigned* __restrict__ barrier) {
    int i = blockIdx.x * blockDim.x + threadIdx.x;
    if (i == 0) *barrier = 0u;
    if (i < SIZE_RES) hbuf[i] = h0[i];
}

// ---------------------------------------------------------------------------
// Phase 1: P[t, j] = sum_k x[t,k] * Win[j,k]   (M=7168, N=4096, K=28)
// one wave per 16x16 output tile, 7 chained V_WMMA_F32_16X16X4_F32
// ---------------------------------------------------------------------------
__global__ void esn_inproj_wmma(const float* __restrict__ x,    // [7168,28]
                                const float* __restrict__ Win,  // [4096,28]
                                float* __restrict__ P) {        // [7168,4096]
    const int lane   = threadIdx.x & 31;
    const int waveId = (blockIdx.x * blockDim.x + threadIdx.x) >> 5;
    const int NT = SIZE_RES / 16;           // 256 N-tiles
    const int tm = waveId >> 8;             // waveId / NT   (448 M-tiles)
    const int tn = waveId & (NT - 1);

    const int m      = tm * 16 + (lane & 15);   // A row for this lane
    const int n      = tn * 16 + (lane & 15);   // B col for this lane
    const int khalf  = (lane >> 4) * 2;         // lanes 16-31 hold K+2,K+3

    v8f c = {};
    #pragma unroll
    for (int k0 = 0; k0 < SIZE_IN; k0 += 4) {   // 28 = 7 * 4 exactly
        const int ka = k0 + khalf;
        v2f a, b;
        a.x = x[m * SIZE_IN + ka];
        a.y = x[m * SIZE_IN + ka + 1];
        b.x = Win[n * SIZE_IN + ka];
        b.y = Win[n * SIZE_IN + ka + 1];
        c = WMMA_F32_16X16X4(a, b, c);
    }
    // D layout: VGPR r -> M = r (+8 for lanes 16-31), N = lane&15
    const int mbase = tm * 16 + ((lane >> 4) << 3);
    const int col   = tn * 16 + (lane & 15);
    #pragma unroll
    for (int r = 0; r < 8; ++r)
        P[(size_t)(mbase + r) * SIZE_RES + col] = c[r];
}

// ---------------------------------------------------------------------------
// Phase 2: persistent sequential scan.
// Each block owns 32 rows of W_res (stays hot in L2). One wave per row:
// lanes split K with float4 loads, shuffle-reduce, tanh, write h_next,
// then device-wide barrier before the next timestep.
// ---------------------------------------------------------------------------
__global__ void esn_scan(const float* __restrict__ x,       // [7168,28]
                         const float* __restrict__ Win,     // [4096,28]
                         const float* __restrict__ Wres,    // [4096,4096]
                         float* __restrict__ hbuf,          // [2][4096]
                         float* __restrict__ h_last,        // [256][4096]
                         const float* __restrict__ in_proj, // [7168,4096] or null
                         unsigned* __restrict__ barrier) {
    __shared__ float4 sh4[SIZE_RES / 4];    // current h (16 KB of 320 KB LDS)
    __shared__ float  sx[32];               // current input row (28 used)

    const int tid  = threadIdx.x;
    const int wave = tid >> 5;
    const int lane = tid & 31;
    const int row0 = blockIdx.x * ROWS_PER_BLK;

    for (int t = 0; t < SEQ; ++t) {
        const float* hcur = hbuf + (size_t)(t & 1) * SIZE_RES;

        // ---- stage h into LDS ----
#if HAVE_ASYNC_LDS
        {
            // direct global -> LDS async copy (ASYNCcnt), 16B per lane
            typedef __attribute__((address_space(1))) v4i* gp_t;
            typedef __attribute__((address_space(3))) v4i* lp_t;
            gp_t gsrc = (gp_t)(void*)const_cast<float*>(hcur);
            lp_t ldst = (lp_t)(void*)sh4;
            #pragma unroll
            for (int i = tid; i < SIZE_RES / 4; i += TPB)       // 4 iterations
                __builtin_amdgcn_global_load_async_to_lds_b128(gsrc + i, ldst + i, 0, 0);
            __builtin_amdgcn_s_wait_asynccnt(0);
        }
#else
        for (int i = tid; i < SIZE_RES / 4; i += TPB)
            sh4[i] = ((const float4*)hcur)[i];
#endif
        if (in_proj == nullptr && tid < SIZE_IN)
            sx[tid] = x[(size_t)t * SIZE_IN + tid];
        __syncthreads();

        float* hnext = hbuf + (size_t)((t + 1) & 1) * SIZE_RES;
        const bool last  = ((t % TIME) == (TIME - 1));
        float* hl = h_last + (size_t)(t / TIME) * SIZE_RES;

        #pragma unroll
        for (int rr = 0; rr < ROWS_PER_WAVE; ++rr) {
            const int r = row0 + wave * ROWS_PER_WAVE + rr;
            const float4* wrow = (const float4*)(Wres + (size_t)r * SIZE_RES);
            // hint the next row's segment for this lane into the caches
            if (rr + 1 < ROWS_PER_WAVE)
                __builtin_prefetch(wrow + (SIZE_RES / 4) + lane, 0, 1);
            float acc = 0.f;
            #pragma unroll 4
            for (int i = lane; i < SIZE_RES / 4; i += 32) {
                float4 w = wrow[i];
                float4 h = sh4[i];
                acc = fmaf(w.x, h.x, acc);
                acc = fmaf(w.y, h.y, acc);
                acc = fmaf(w.z, h.z, acc);
                acc = fmaf(w.w, h.w, acc);
            }
            #pragma unroll
            for (int off = 16; off > 0; off >>= 1)
                acc += __shfl_xor(acc, off, 32);
            if (lane == 0) {
                float up;
                if (in_proj) {
                    up = in_proj[(size_t)t * SIZE_RES + r];
                } else {
                    up = 0.f;
                    #pragma unroll
                    for (int i = 0; i < SIZE_IN; ++i)
                        up = fmaf(Win[r * SIZE_IN + i], sx[i], up);
                }
                const float hv = tanhf(up + acc);
                hnext[r] = hv;
                if (last) hl[r] = hv;
            }
        }

        // ---- device-wide barrier (monotonic counter, no reset races) ----
        __threadfence();            // release: flush this block's h writes
        __syncthreads();
        if (tid == 0) {
            __hip_atomic_fetch_add(barrier, 1u, __ATOMIC_RELEASE,
                                   __HIP_MEMORY_SCOPE_AGENT);
            const unsigned target = (unsigned)(t + 1) * (unsigned)NBLK;
            while (__hip_atomic_load(barrier, __ATOMIC_ACQUIRE,
                                     __HIP_MEMORY_SCOPE_AGENT) < target)
                __builtin_amdgcn_s_sleep(1);
        }
        __syncthreads();
        __threadfence();            // acquire: invalidate stale L0 lines
    }
}

// ---------------------------------------------------------------------------
// Phase 3: out[i,o] = sum_k h_last[i,k] * Wout[o,k]  (M=256, N=10->16, K=4096)
// 16 waves, one 16x16 tile each, 1024 chained V_WMMA_F32_16X16X4_F32
// ---------------------------------------------------------------------------
__global__ void esn_readout_wmma(const float* __restrict__ h_last, // [256,4096]
                                 const float* __restrict__ Wout,   // [10,4096]
                                 float* __restrict__ out) {        // [256,10]
    const int wave  = threadIdx.x >> 5;      // 0..15 -> M tile
    const int lane  = threadIdx.x & 31;
    const int i0    = wave * 16;
    const int m     = i0 + (lane & 15);
    const int n     = lane & 15;
    const int khalf = (lane >> 4) * 2;
    const bool nok  = (n < SIZE_OUT);

    v8f c = {};
    for (int k0 = 0; k0 < SIZE_RES; k0 += 4) {
        const int ka = k0 + khalf;
        v2f a, b;
        a.x = h_last[(size_t)m * SIZE_RES + ka];
        a.y = h_last[(size_t)m * SIZE_RES + ka + 1];
        b.x = nok ? Wout[n * SIZE_RES + ka]     : 0.f;
        b.y = nok ? Wout[n * SIZE_RES + ka + 1] : 0.f;
        c = WMMA_F32_16X16X4(a, b, c);
    }
    const int mbase = i0 + ((lane >> 4) << 3);
    #pragma unroll
    for (int r = 0; r < 8; ++r)
        if (nok) out[(mbase + r) * SIZE_OUT + n] = c[r];
}

// ---------------------------------------------------------------------------
extern "C" void kernel_launch(void* const* d_in, const int* in_sizes, int n_in,
                              void* d_out, int out_size, void* d_ws, size_t ws_size,
                              hipStream_t stream) {
    const float* x    = (const float*)d_in[0];   // [256,28,28]
    const float* Win  = (const float*)d_in[1];   // [4096,28]
    const float* Wres = (const float*)d_in[2];   // [4096,4096]
    const float* Wout = (const float*)d_in[3];   // [10,4096]
    const float* h0   = (const float*)d_in[4];   // [4096]
    float* out = (float*)d_out;

    char* ws = (char*)d_ws;
    unsigned* barrier = (unsigned*)ws;                          // 256 B slot
    float* hbuf   = (float*)(ws + 256);                         // 32 KB
    float* h_last = (float*)(ws + 256 + 2 * SIZE_RES * 4);      // 4 MB
    const size_t base    = 256 + 2 * SIZE_RES * 4 + (size_t)BATCH * SIZE_RES * 4;
    const size_t npBytes = (size_t)SEQ * SIZE_RES * 4;          // 117 MB
    float* in_proj = (ws_size >= base + npBytes) ? (float*)(ws + base) : nullptr;

    esn_init<<<(SIZE_RES + 255) / 256, 256, 0, stream>>>(h0, hbuf, barrier);

    if (in_proj) {
        // 448 * 256 tiles / 8 waves per block
        const int tiles  = (SEQ / 16) * (SIZE_RES / 16);
        const int blocks = tiles / WAVES_PER_BLK;               // 14336
        esn_inproj_wmma<<<blocks, TPB, 0, stream>>>(x, Win, in_proj);
    }

    esn_scan<<<NBLK, TPB, 0, stream>>>(x, Win, Wres, hbuf, h_last, in_proj, barrier);

    esn_readout_wmma<<<1, 512, 0, stream>>>(h_last, Wout, out);
}